// SSIM_15925738734266
// MI455X (gfx1250) — compile-verified
//
#include <hip/hip_runtime.h>

typedef __attribute__((ext_vector_type(2))) float v2f;
typedef __attribute__((ext_vector_type(8))) float v8f;

#define IMG_H 512
#define IMG_W 512
#define TILE 32
#define SLABW 44   // 32 + 10 halo + 2 pad (K padded 26->28)
#define SLABH 48   // 32 + 10 halo + 6 pad (M padded 26->32 on 2nd rowtile)
#define SPITCH 20  // staging pitch: conflict-free for D-store and B-read patterns

__global__ __launch_bounds__(128) void ssim_wmma_tile_kernel(
    const float* __restrict__ X, const float* __restrict__ Y,
    const float* __restrict__ Wwin, float* __restrict__ partial)
{
    __shared__ float sG[16];
    __shared__ float sX[SLABH * SLABW];
    __shared__ float sY[SLABH * SLABW];
    __shared__ float sStage[4][32 * SPITCH];
    __shared__ float sRed[4];

    const int tid  = threadIdx.x;
    const int lane = tid & 31;
    const int wav  = tid >> 5;
    const int half = lane >> 4;   // which 16-lane half
    const int l16  = lane & 15;

    // block -> (image, 32x32 tile); 256 tiles per 512x512 image
    const int img = blockIdx.x >> 8;
    const int t   = blockIdx.x & 255;
    const int by  = (t >> 4) * TILE;
    const int bx  = (t & 15) * TILE;

    // Recover the 1-D gaussian from the 2-D window: g[i] = sum_j w[i][j]
    if (tid < 16) {
        float s = 0.f;
        if (tid < 11) {
            #pragma unroll
            for (int j = 0; j < 11; ++j) s += Wwin[tid * 11 + j];
        }
        sG[tid] = s;
    }

    // Load zero-padded x/y slabs (halo of 5 each side, padded to 48x44)
    const float* Xi = X + (size_t)img * (IMG_H * IMG_W);
    const float* Yi = Y + (size_t)img * (IMG_H * IMG_W);
    for (int idx = tid; idx < SLABH * SLABW; idx += 128) {
        const int r  = idx / SLABW;
        const int c  = idx % SLABW;
        const int gy = by - 5 + r;
        const int gx = bx - 5 + c;
        const bool ok = (gy >= 0) && (gy < IMG_H) && (gx >= 0) && (gx < IMG_W);
        const int g  = gy * IMG_W + gx;
        sX[idx] = ok ? Xi[g] : 0.f;
        sY[idx] = ok ? Yi[g] : 0.f;
    }
    __syncthreads();

    // Per-lane banded-Gaussian fragments. The same per-lane values serve as
    // the B-fragment of W[k][n]=g[k-n] (horizontal) and the A-fragment of
    // G[m][j]=g[j-m] (vertical), because G = W^T and A(lane=row)/B(lane=col)
    // layouts mirror each other.
    v2f Wc[7];
    #pragma unroll
    for (int c = 0; c < 7; ++c) {
        const int kk = 4 * c + 2 * half;
        const int d0 = kk - l16;
        const int d1 = kk + 1 - l16;
        v2f w;
        w[0] = (d0 >= 0 && d0 <= 10) ? sG[d0] : 0.f;
        w[1] = (d1 >= 0 && d1 <= 10) ? sG[d1] : 0.f;
        Wc[c] = w;
    }

    // Each wave owns one 16x16 sub-tile of the 32x32 block tile
    const int sy = (wav >> 1) * 16;
    const int sx = (wav & 1) * 16;

    // ---------- Horizontal pass: TMP = IN x W, 5 planes, 2 row-tiles ----------
    v8f accH[2][5];
    #pragma unroll
    for (int rt = 0; rt < 2; ++rt)
        #pragma unroll
        for (int p = 0; p < 5; ++p) accH[rt][p] = {};

    #pragma unroll
    for (int rt = 0; rt < 2; ++rt) {
        const int row = sy + 16 * rt + l16;            // A-matrix row for this lane
        #pragma unroll
        for (int c = 0; c < 7; ++c) {
            const int col = sx + 4 * c + 2 * half;     // K position pair
            const float* px = &sX[row * SLABW + col];
            const float* py = &sY[row * SLABW + col];
            v2f xa, ya;
            xa[0] = px[0]; xa[1] = px[1];
            ya[0] = py[0]; ya[1] = py[1];
            const v2f xx = xa * xa;
            const v2f yy = ya * ya;
            const v2f xy = xa * ya;
            accH[rt][0] = __builtin_amdgcn_wmma_f32_16x16x4_f32(false, xa, false, Wc[c], (short)0, accH[rt][0], false, false);
            accH[rt][1] = __builtin_amdgcn_wmma_f32_16x16x4_f32(false, ya, false, Wc[c], (short)0, accH[rt][1], false, false);
            accH[rt][2] = __builtin_amdgcn_wmma_f32_16x16x4_f32(false, xx, false, Wc[c], (short)0, accH[rt][2], false, false);
            accH[rt][3] = __builtin_amdgcn_wmma_f32_16x16x4_f32(false, yy, false, Wc[c], (short)0, accH[rt][3], false, false);
            accH[rt][4] = __builtin_amdgcn_wmma_f32_16x16x4_f32(false, xy, false, Wc[c], (short)0, accH[rt][4], false, false);
        }
    }

    // ---------- Vertical pass: OUT = G x TMP per plane (via LDS staging) ----------
    v8f fin[5];
    float* stg = &sStage[wav][0];
    #pragma unroll
    for (int p = 0; p < 5; ++p) {
        // Store both row-tiles of TMP (D layout: lane=col n, vgpr v -> row v+8*half)
        #pragma unroll
        for (int v = 0; v < 8; ++v) {
            stg[(v + 8 * half) * SPITCH + l16]      = accH[0][p][v];
            stg[(16 + v + 8 * half) * SPITCH + l16] = accH[1][p][v];
        }
        __syncthreads();
        v8f acc = {};
        #pragma unroll
        for (int c = 0; c < 7; ++c) {
            const int jj = 4 * c + 2 * half;           // K row pair of TMP
            v2f b;
            b[0] = stg[jj * SPITCH + l16];
            b[1] = stg[(jj + 1) * SPITCH + l16];
            acc = __builtin_amdgcn_wmma_f32_16x16x4_f32(false, Wc[c], false, b, (short)0, acc, false, false);
        }
        fin[p] = acc;
        __syncthreads();
    }

    // ---------- SSIM map + reduction ----------
    const float C1v = 0.0001f;   // 0.01^2
    const float C2v = 0.0009f;   // 0.03^2
    const float EPSv = 1e-8f;
    float lsum = 0.f;
    #pragma unroll
    for (int v = 0; v < 8; ++v) {
        const float mux = fin[0][v], muy = fin[1][v];
        const float ex2 = fin[2][v], ey2 = fin[3][v], exy = fin[4][v];
        const float mux2 = mux * mux, muy2 = muy * muy, muxy = mux * muy;
        const float sx2 = ex2 - mux2;
        const float sy2 = ey2 - muy2;
        const float sxy = exy - muxy;
        const float num = (2.f * muxy + C1v) * (2.f * sxy + C2v);
        const float den = (mux2 + muy2 + C1v) * (sx2 + sy2 + C2v);
        float r = num / (den + EPSv);
        r = fminf(fmaxf(r, 0.f), 1.f);
        lsum += r;
    }
    #pragma unroll
    for (int off = 16; off > 0; off >>= 1)
        lsum += __shfl_down(lsum, off, 32);
    if (lane == 0) sRed[wav] = lsum;
    __syncthreads();
    if (tid == 0)
        partial[blockIdx.x] = sRed[0] + sRed[1] + sRed[2] + sRed[3];
}

__global__ __launch_bounds__(256) void ssim_reduce_kernel(
    const float* __restrict__ partial, int n, float* __restrict__ out, double invCount)
{
    __shared__ double sd[256];
    double s = 0.0;
    for (int i = threadIdx.x; i < n; i += 256) s += (double)partial[i];
    sd[threadIdx.x] = s;
    __syncthreads();
    for (int off = 128; off > 0; off >>= 1) {
        if (threadIdx.x < off) sd[threadIdx.x] += sd[threadIdx.x + off];
        __syncthreads();
    }
    if (threadIdx.x == 0) out[0] = (float)(sd[0] * invCount);
}

extern "C" void kernel_launch(void* const* d_in, const int* in_sizes, int n_in,
                              void* d_out, int out_size, void* d_ws, size_t ws_size,
                              hipStream_t stream) {
    const float* x = (const float*)d_in[0];
    const float* y = (const float*)d_in[1];
    const float* w = (const float*)d_in[2];   // [1,1,11,11]
    float* out = (float*)d_out;
    float* partial = (float*)d_ws;

    const int N = in_sizes[0] / (IMG_H * IMG_W);   // 32 images
    const int nblocks = N * 256;                   // 256 tiles of 32x32 per image

    ssim_wmma_tile_kernel<<<nblocks, 128, 0, stream>>>(x, y, w, partial);
    ssim_reduce_kernel<<<1, 256, 0, stream>>>(
        partial, nblocks, out, 1.0 / ((double)N * IMG_H * IMG_W));
}